// ModalityEncoder_56650618635007
// MI455X (gfx1250) — compile-verified
//
#include <hip/hip_runtime.h>
#include <math.h>

// MI455X / gfx1250, wave32. Fused Clifford-MLP + rotor.
// GEMMs on V_WMMA_F32_16X16X4_F32 (full fp32, XDL-rate).
// Norm reductions on DPP8 / DPP16 (pure VALU).
// Branch-free erf via rcp/exp2 transcendentals; rsq for all norms.
// C-layout -> A-layout transposes via per-wave LDS tiles.

typedef __attribute__((ext_vector_type(2))) float v2f;
typedef __attribute__((ext_vector_type(4))) float v4f;
typedef __attribute__((ext_vector_type(8))) float v8f;

#define WAVES_PER_BLOCK 8

// DPP8 selector: lane i reads lane sel[i] within its 8-lane group
#define DPP8SEL(a,b,c,d,e,f,g,h) \
  ((a)|((b)<<3)|((c)<<6)|((d)<<9)|((e)<<12)|((f)<<15)|((g)<<18)|((h)<<21))
#define DPP8_XOR1 DPP8SEL(1,0,3,2,5,4,7,6)
#define DPP8_XOR2 DPP8SEL(2,3,0,1,6,7,4,5)
#define DPP8_XOR4 DPP8SEL(4,5,6,7,0,1,2,3)
#define DPP16_ROW_ROR8 0x128   // rotate 16-lane row by 8 == xor-8 swap

// sum over each group of 8 contiguous lanes (VALU-only butterfly)
__device__ __forceinline__ float sum8(float s) {
  s += __int_as_float(__builtin_amdgcn_mov_dpp8(__float_as_int(s), DPP8_XOR1));
  s += __int_as_float(__builtin_amdgcn_mov_dpp8(__float_as_int(s), DPP8_XOR2));
  s += __int_as_float(__builtin_amdgcn_mov_dpp8(__float_as_int(s), DPP8_XOR4));
  return s;
}
// add value from the other 8-lane half of the 16-lane row
__device__ __forceinline__ float swap8_add(float s) {
  return s + __int_as_float(__builtin_amdgcn_mov_dpp(
      __float_as_int(s), DPP16_ROW_ROR8, 0xf, 0xf, true));
}

// Branch-free erf for x >= 0 (Abramowitz–Stegun 7.1.26, |err| < 1.5e-7).
// Uses v_rcp_f32 + v_exp_f32; no exec-mask divergence.
__device__ __forceinline__ float erf_pos(float xx) {
  float t = __builtin_amdgcn_rcpf(fmaf(0.3275911f, xx, 1.0f));
  float p = fmaf(1.061405429f, t, -1.453152027f);
  p = fmaf(p, t, 1.421413741f);
  p = fmaf(p, t, -0.284496736f);
  p = fmaf(p, t, 0.254829592f);
  p = p * t;
  float e = __builtin_amdgcn_exp2f(-1.4426950408889634f * xx * xx);
  return fmaf(-p, e, 1.0f);
}

// Cl(3,0,0) Cayley: blades ordered [1,e1,e2,e3,e12,e13,e23,e123]
// index<->mask map is the involution {0,1,2,4,3,5,6,7}.
__device__ __forceinline__ void cay(int i, int j, int& k, float& sg) {
  const int bl[8] = {0, 1, 2, 4, 3, 5, 6, 7};
  int a = bl[i], b = bl[j];
  int s = 0, aa = a >> 1;
  while (aa) { s += __popc(aa & b); aa >>= 1; }
  k  = bl[a ^ b];                 // inverse map equals bl
  sg = (s & 1) ? -1.0f : 1.0f;
}

__global__ __launch_bounds__(256)
void clifford_rotor_fused(const float* __restrict__ x,
                          const float* __restrict__ W1,
                          const float* __restrict__ b1,
                          const float* __restrict__ g1,
                          const float* __restrict__ W2,
                          const float* __restrict__ b2,
                          const float* __restrict__ g2,
                          const float* __restrict__ biv,
                          float* __restrict__ out, int B) {
  // ---- LDS: fused weight matrices + per-wave transpose tiles ----
  __shared__ __align__(16) float lA1[8 * 32];    // GEMM1 weights [K=8][N=32]
  __shared__ __align__(16) float lA2[32 * 16];   // GEMM2 weights [K=32][N=16] (N 8..15 = A2*Arot)
  __shared__ float lP1[64], lP2[64], lArot[64];
  __shared__ float lB1[32], lB2[16], lG1[4], lG2[1];
  __shared__ __align__(16) float lT[WAVES_PER_BLOCK * 512];  // 16x32 f32 tile per wave

  const int tid = threadIdx.x;

  // Rotor R = exp(Bv): scalar + bivector part (redundant per thread, tiny)
  const float bv0 = biv[0], bv1 = biv[1], bv2 = biv[2];
  const float th  = sqrtf(bv0 * bv0 + bv1 * bv1 + bv2 * bv2 + 1e-12f);
  const float snc = sinf(th) / th, ct = cosf(th);
  float R[8]  = {ct, 0, 0, 0,  snc * bv0,  snc * bv1,  snc * bv2, 0};
  float Rr[8] = {ct, 0, 0, 0, -snc * bv0, -snc * bv1, -snc * bv2, 0};

  // ---- Phase 1: A1, P1/P2 (rotor halves), bias/gamma copies ----
  for (int e = tid; e < 256; e += blockDim.x) {   // A1[l][n], n = o*8+k
    int l = e >> 5, n = e & 31, o = n >> 3, k = n & 7;
    float s = 0.f;
    for (int j = 0; j < 8; ++j) { int kk; float sg; cay(j, l, kk, sg);
      if (kk == k) s += sg * W1[j * 4 + o]; }
    lA1[e] = s;
  }
  for (int e = tid; e < 64; e += blockDim.x) {
    int a = e >> 3, k = e & 7;
    float s1 = 0.f, s2 = 0.f;
    for (int i = 0; i < 8; ++i) { int kk; float sg;
      cay(i, a, kk, sg); if (kk == k) s1 += sg * R[i];   // P1[a][k]=sum_i R_i C[i,a,k]
      cay(a, i, kk, sg); if (kk == k) s2 += sg * Rr[i];  // P2[a][k]=sum_i Rr_i C[a,i,k]
    }
    lP1[e] = s1; lP2[e] = s2;
  }
  for (int e = tid; e < 32; e += blockDim.x) lB1[e] = b1[e];
  if (tid < 4) lG1[tid] = g1[tid];
  if (tid == 0) lG2[0] = g2[0];
  __syncthreads();

  // ---- Phase 2: Arot = P1 * P2 ----
  for (int e = tid; e < 64; e += blockDim.x) {
    int j = e >> 3, m = e & 7; float s = 0.f;
    for (int k = 0; k < 8; ++k) s += lP1[j * 8 + k] * lP2[k * 8 + m];
    lArot[e] = s;
  }
  __syncthreads();

  // ---- Phase 3: A2ext = [A2 | A2*Arot], bias2ext = [b2 | b2*Arot] ----
  for (int e = tid; e < 512; e += blockDim.x) {
    int m = e >> 4, n = e & 15, ic = m >> 3, l = m & 7;
    float s = 0.f;
    if (n < 8) {
      for (int j = 0; j < 8; ++j) { int kk; float sg; cay(j, l, kk, sg);
        if (kk == n) s += sg * W2[j * 4 + ic]; }
    } else {
      for (int k = 0; k < 8; ++k) {
        float a2 = 0.f;
        for (int j = 0; j < 8; ++j) { int kk; float sg; cay(j, l, kk, sg);
          if (kk == k) a2 += sg * W2[j * 4 + ic]; }
        s += a2 * lArot[k * 8 + (n - 8)];
      }
    }
    lA2[e] = s;
  }
  if (tid < 16) {
    float s;
    if (tid < 8) s = b2[tid];
    else { s = 0.f; for (int k = 0; k < 8; ++k) s += b2[k] * lArot[k * 8 + tid - 8]; }
    lB2[tid] = s;
  }
  __syncthreads();

  // ---- Per-wave setup ----
  const int lane = tid & 31, wave = tid >> 5;
  const int half = lane >> 4, li = lane & 15;   // wave32: half-selects + 16 lanes
  float* T = lT + wave * 512;

  // B-operands in registers (VGPR j of K-chunk c holds K = 4c + j + 2*half, N = li)
  v2f w1op[2][2];
#pragma unroll
  for (int t = 0; t < 2; ++t)
#pragma unroll
    for (int c = 0; c < 2; ++c) {
      int k0 = 4 * c + 2 * half;
      w1op[t][c].x = lA1[(k0    ) * 32 + t * 16 + li];
      w1op[t][c].y = lA1[(k0 + 1) * 32 + t * 16 + li];
    }
  v2f w2op[8];
#pragma unroll
  for (int c = 0; c < 8; ++c) {
    int k0 = 4 * c + 2 * half;
    w2op[c].x = lA2[(k0    ) * 16 + li];
    w2op[c].y = lA2[(k0 + 1) * 16 + li];
  }
  const float ga = lG1[li >> 3], gb = lG1[2 + (li >> 3)], g2v = lG2[0];
  v8f bias0, bias1, bias2;
#pragma unroll
  for (int r = 0; r < 8; ++r) {
    bias0[r] = lB1[li]; bias1[r] = lB1[16 + li]; bias2[r] = lB2[li];
  }

  const int tiles    = (B + 15) >> 4;
  const int nw       = gridDim.x * WAVES_PER_BLOCK;
  const int stepRows = nw * 16;

  for (int tile = blockIdx.x * WAVES_PER_BLOCK + wave; tile < tiles; tile += nw) {
    const int rowBase = tile << 4;           // < 2^21, 32-bit safe

    // Prefetch this wave's next tile (global_prefetch_b8), 64B/lane covers 2KiB
    if (rowBase + stepRows + 16 <= B)
      __builtin_prefetch(x + (unsigned)(rowBase + stepRows) * 8u + (unsigned)lane * 16u, 0, 0);

    // A-operand: 16x8 x-tile as two 16x4 K-chunks (K = 2*half + {0,1} per VGPR pair)
    int rA = rowBase + li; rA = (rA < B) ? rA : (B - 1);   // v_min_i32, branchless
    const float* xr = x + ((unsigned)rA << 3) + (unsigned)(2 * half);
    v2f a0 = *(const v2f*)(xr);
    v2f a1 = *(const v2f*)(xr + 4);

    // ---- GEMM1: [16x8] x [8x32] + b1, two N-tiles, K in two chunks ----
    v8f c0 = bias0, c1 = bias1;
    c0 = __builtin_amdgcn_wmma_f32_16x16x4_f32(false, a0, false, w1op[0][0], (short)0, c0, false, false);
    c0 = __builtin_amdgcn_wmma_f32_16x16x4_f32(false, a1, false, w1op[0][1], (short)0, c0, false, false);
    c1 = __builtin_amdgcn_wmma_f32_16x16x4_f32(false, a0, false, w1op[1][0], (short)0, c1, false, false);
    c1 = __builtin_amdgcn_wmma_f32_16x16x4_f32(false, a1, false, w1op[1][1], (short)0, c1, false, false);

    // ---- channel RMS-LN + magnitude-gated GELU (DPP8 butterflies) ----
    auto lngelu = [&](v8f& cc, float g) {
#pragma unroll
      for (int r = 0; r < 8; ++r) {
        float v   = cc[r];
        float s   = sum8(v * v);                               // sum over 8 blades
        float inv = __builtin_amdgcn_rsqf(0.125f * s + 1e-6f); // 1/rms (v_rsq_f32)
        float hn  = __builtin_amdgcn_sqrtf((g * g) * s * inv * inv + 1e-6f);
        float gate = fmaf(0.5f, erf_pos(hn * 0.70710678118654752f), 0.5f);
        cc[r] = (g * gate * inv) * v;
      }
    };
    lngelu(c0, ga);
    lngelu(c1, gb);

    // ---- transpose C-layout -> A-layout through per-wave LDS tile ----
#pragma unroll
    for (int r = 0; r < 8; ++r) {
      int row = r + 8 * half;
      T[row * 32 + li]      = c0[r];
      T[row * 32 + 16 + li] = c1[r];
    }
    v2f a2[8];
#pragma unroll
    for (int c = 0; c < 8; ++c)
      a2[c] = *(const v2f*)&T[li * 32 + 4 * c + 2 * half];

    // ---- GEMM2 (+ fused rotor in N=8..15): [16x32] x [32x16] + bias2ext ----
    v8f cc2 = bias2;
#pragma unroll
    for (int c = 0; c < 8; ++c)
      cc2 = __builtin_amdgcn_wmma_f32_16x16x4_f32(false, a2[c], false, w2op[c], (short)0, cc2, false, false);

    // ---- final LN (norm over cols 0..7) scales rotor cols 8..15 ----
    float o[8];
#pragma unroll
    for (int r = 0; r < 8; ++r) {
      float v = cc2[r];
      float s = (li < 8) ? v * v : 0.f;
      s = sum8(s);
      s = swap8_add(s);                  // broadcast y2-norm to rotor lanes
      o[r] = g2v * v * __builtin_amdgcn_rsqf(0.125f * s + 1e-6f);
    }
    if (li >= 8) {                       // single exec toggle for all 8 stores
      float* Tr = T + 8 * half * 8 + (li - 8);
#pragma unroll
      for (int r = 0; r < 8; ++r) Tr[r * 8] = o[r];
    }

    // ---- repack + one coalesced b128 store per lane (512B / tile) ----
    v4f ov = *(const v4f*)&T[lane * 4];
    int orow = rowBase + (lane >> 1);
    if (orow < B)
      *(v4f*)(out + (unsigned)rowBase * 8u + (unsigned)lane * 4u) = ov;
  }
}

extern "C" void kernel_launch(void* const* d_in, const int* in_sizes, int n_in,
                              void* d_out, int out_size, void* d_ws, size_t ws_size,
                              hipStream_t stream) {
  const float* x   = (const float*)d_in[0];
  const float* W1  = (const float*)d_in[1];
  const float* b1  = (const float*)d_in[2];
  const float* g1  = (const float*)d_in[3];
  const float* W2  = (const float*)d_in[4];
  const float* b2  = (const float*)d_in[5];
  const float* g2  = (const float*)d_in[6];
  const float* biv = (const float*)d_in[7];
  float* out = (float*)d_out;

  int B = in_sizes[0] / 8;
  int tiles = (B + 15) / 16;
  int blocks = (tiles + WAVES_PER_BLOCK - 1) / WAVES_PER_BLOCK;
  if (blocks > 2048) blocks = 2048;   // persistent waves, >=8 tiles each at B=2^21
  if (blocks < 1) blocks = 1;

  clifford_rotor_fused<<<blocks, 256, 0, stream>>>(x, W1, b1, g1, W2, b2, g2, biv, out, B);
}